// ProbAttention_40999757808016
// MI455X (gfx1250) — compile-verified
//
#include <hip/hip_runtime.h>
#include <hip/hip_bf16.h>

// ProbSparse attention (Informer): B=4, L=S=2048, H=8, D=64, n_top=sample_k=40.
// fp32 end-to-end; GEMMs on V_WMMA_F32_16X16X4_F32; K/V tiles staged into LDS
// once per block by the Tensor Data Mover (tensor_load_to_lds + TENSORcnt),
// with TDM LDS padding (row stride 68 floats) for bank-conflict-free reads.
// Flash-decoding split over S (8 splits/(b,h)) + combine; softmax denominator
// computed on the matrix unit (P @ ones).

typedef float v2f __attribute__((ext_vector_type(2)));
typedef float v8f __attribute__((ext_vector_type(8)));
typedef unsigned int v4u __attribute__((ext_vector_type(4)));
typedef int v4i __attribute__((ext_vector_type(4)));
typedef int v8i __attribute__((ext_vector_type(8)));

#define B_  4
#define L_  2048
#define S_  2048
#define H_  8
#define D_  64
#define NTOP 40
#define SAMPK 40
#define NTILES 3          // ceil(40/16) row tiles of reduced queries (48 padded rows)
#define NROWS 48
#define NSPLIT 8          // S partitions per (b,h)
#define SPB (S_ / NSPLIT) // 256 keys per block
#define SB 64             // keys per inner iteration
#define KSTR 68           // LDS row stride in floats (64 + 4 TDM pad DWORDs)
#define SCALE 0.125f      // 1/sqrt(64)

__device__ __forceinline__ v8f wmma_f32(v2f a, v2f b, v8f c) {
    return __builtin_amdgcn_wmma_f32_16x16x4_f32(
        /*neg_a=*/false, a, /*neg_b=*/false, b,
        /*c_mod=*/(short)0, c, /*reuse_a=*/false, /*reuse_b=*/false);
}

// TDM: DMA a 64x64 f32 tile (global row stride 512 floats = H*D) into LDS with
// 4-DWORD padding after every 64 DWORDs (LDS row stride = 68 floats).
__device__ __forceinline__ void tdm_load_tile(unsigned int lds_off, const float* gptr) {
    unsigned long long ga = (unsigned long long)(uintptr_t)gptr;
    v4u g0;
    g0.x = 1u;                                       // count=1, user descriptor
    g0.y = lds_off;                                  // lds_addr (bytes)
    g0.z = (unsigned int)(ga & 0xffffffffu);         // global_addr[31:0]
    g0.w = (unsigned int)((ga >> 32) & 0x1ffffffu)   // global_addr[56:32]
         | (2u << 30);                               // type = 2 ("image")
    v8i g1;
    g1[0] = (int)((2u << 16)      // data_size = 4B
                | (1u << 20)      // pad_enable
                | (5u << 22)      // pad_interval: 64 DWORDs
                | (3u << 25));    // pad_amount: 4 DWORDs
    g1[1] = (int)(64u << 16);     // tensor_dim0[15:0] = 64
    g1[2] = (int)(2048u << 16);   // tensor_dim1[15:0] = 2048
    g1[3] = (int)(64u << 16);     // tile_dim0 = 64
    g1[4] = 64;                   // tile_dim1 = 64 (tile_dim2 = 0)
    g1[5] = 512;                  // tensor_dim0_stride = 512 elements
    g1[6] = 0;                    // dim0_stride hi / dim1_stride lo
    g1[7] = 0;                    // dim1_stride hi
    v4i z4 = {0, 0, 0, 0};
#if __clang_major__ >= 23
    v8i z8 = {0, 0, 0, 0, 0, 0, 0, 0};
    __builtin_amdgcn_tensor_load_to_lds(g0, g1, z4, z4, z8, 0);
#else
    __builtin_amdgcn_tensor_load_to_lds(g0, g1, z4, z4, 0);
#endif
}

// ---------------------------------------------------------------------------
// Kernel 1: M[b,h,l] = max_u(q.k_samp) - sum_u(q.k_samp)/S   (gathered dots)
// ---------------------------------------------------------------------------
__global__ void probattn_measure(const float* __restrict__ q,
                                 const float* __restrict__ k,
                                 const int*   __restrict__ idx,
                                 float* __restrict__ M) {
    int gid = blockIdx.x * blockDim.x + threadIdx.x;   // (b*H+h)*L + l
    if (gid >= B_ * H_ * L_) return;
    int l  = gid % L_;
    int bh = gid / L_;
    int h  = bh % H_;
    int b  = bh / H_;

    const float4* qrow = (const float4*)(q + (((size_t)b * L_ + l) * H_ + h) * D_);
    float4 qv[16];
#pragma unroll
    for (int j = 0; j < 16; ++j) qv[j] = qrow[j];

    float mx  = -__builtin_inff();
    float sum = 0.0f;
    for (int u = 0; u < SAMPK; ++u) {
        int s = idx[l * SAMPK + u];
        const float4* krow = (const float4*)(k + (((size_t)b * S_ + s) * H_ + h) * D_);
        float dot = 0.0f;
#pragma unroll
        for (int j = 0; j < 16; ++j) {
            float4 kv = krow[j];
            dot += qv[j].x * kv.x + qv[j].y * kv.y + qv[j].z * kv.z + qv[j].w * kv.w;
        }
        mx = fmaxf(mx, dot);
        sum += dot;
    }
    M[gid] = mx - sum * (1.0f / (float)S_);
}

// ---------------------------------------------------------------------------
// Kernel 2: per-(b,h) top-40 argmax over L=2048 (iterative; destroys M)
// ---------------------------------------------------------------------------
__global__ void probattn_topk(float* __restrict__ M, int* __restrict__ mtop) {
    int bh = blockIdx.x;
    float* Mp = M + (size_t)bh * L_;
    __shared__ float sval[256];
    __shared__ int   sidx[256];
    int tid = threadIdx.x;

    for (int iter = 0; iter < NTOP; ++iter) {
        float best = -__builtin_inff();
        int   bidx = 0;
        for (int i = tid; i < L_; i += 256) {
            float v = Mp[i];
            if (v > best) { best = v; bidx = i; }
        }
        sval[tid] = best; sidx[tid] = bidx;
        __syncthreads();
        for (int off = 128; off > 0; off >>= 1) {
            if (tid < off) {
                if (sval[tid + off] > sval[tid] ||
                    (sval[tid + off] == sval[tid] && sidx[tid + off] < sidx[tid])) {
                    sval[tid] = sval[tid + off];
                    sidx[tid] = sidx[tid + off];
                }
            }
            __syncthreads();
        }
        if (tid == 0) {
            mtop[bh * NTOP + iter] = sidx[0];
            Mp[sidx[0]] = -__builtin_inff();
        }
        __syncthreads();
    }
}

// ---------------------------------------------------------------------------
// Kernel 3: vmean[b,h,d] = mean_s v[b,s,h,d]
// ---------------------------------------------------------------------------
__global__ void probattn_vmean(const float* __restrict__ v, float* __restrict__ vmean) {
    int bh = blockIdx.x;
    int b = bh / H_, h = bh % H_;
    int d = threadIdx.x;
    const float* vp = v + (((size_t)b * S_) * H_ + h) * D_ + d;
    float s = 0.0f;
    for (int ss = 0; ss < S_; ++ss) s += vp[(size_t)ss * H_ * D_];
    vmean[(size_t)bh * D_ + d] = s * (1.0f / (float)S_);
}

// ---------------------------------------------------------------------------
// Kernel 4: out[b,l,h,d] = vmean[b,h,d]  (default context)
// ---------------------------------------------------------------------------
__global__ void probattn_bcast(const float* __restrict__ vmean, float* __restrict__ out) {
    int gid = blockIdx.x * blockDim.x + threadIdx.x;
    if (gid >= B_ * L_ * H_ * D_) return;
    int d = gid % D_;
    int h = (gid / D_) % H_;
    int b = gid / (L_ * H_ * D_);
    out[gid] = vmean[((size_t)b * H_ + h) * D_ + d];
}

// ---------------------------------------------------------------------------
// Kernel 5: split-S flash attention over the 40 selected queries.
// Block = (bh, split); 3 waves, wave w owns reduced-query row tile w.
// Per 64-key iteration: TDM stages K/V 64x64 tiles into LDS (once per block),
// then 64 score WMMAs, 64 P@V WMMAs, 16 P@ones WMMAs (online denominator).
// Emits per-row partials (m, l, unnormalized O) to workspace.
// ---------------------------------------------------------------------------
__global__ void __launch_bounds__(96)
probattn_split(const float* __restrict__ q,
               const float* __restrict__ k,
               const float* __restrict__ v,
               const int*   __restrict__ mtop,
               float* __restrict__ part_acc,   // [B*H][NSPLIT][NROWS][D]
               float* __restrict__ part_m,     // [B*H][NSPLIT][NROWS]
               float* __restrict__ part_l) {   // [B*H][NSPLIT][NROWS]
    __shared__ float ktile[SB * KSTR];         // 64 key rows, stride 68
    __shared__ float vtile[SB * KSTR];         // 64 value rows, stride 68
    __shared__ float pbuf[NTILES][16 * KSTR];  // P staging (C/D -> A layout)

    int bh    = blockIdx.x / NSPLIT;
    int split = blockIdx.x % NSPLIT;
    int b = bh / H_, h = bh % H_;
    int wave = threadIdx.x >> 5;
    int lane = threadIdx.x & 31;
    int half = lane >> 4;        // 0: K{0,1} slot, 1: K{2,3} slot
    int ln   = lane & 15;        // row (A) / col (B,D) index

    unsigned int koff = (unsigned int)(uintptr_t)(void*)&ktile[0];
    unsigned int voff = (unsigned int)(uintptr_t)(void*)&vtile[0];

    // ---- load this wave's Q-tile rows in A-operand order --------------------
    int n_a   = wave * 16 + ln;              // reduced-query index of A row
    bool aval = n_a < NTOP;
    int lq    = aval ? mtop[bh * NTOP + n_a] : 0;
    const float* qrow = q + (((size_t)b * L_ + lq) * H_ + h) * D_;

    v2f aq[16];
#pragma unroll
    for (int kk = 0; kk < 16; ++kk) {
        int d0 = kk * 4 + half * 2;
        if (aval) { aq[kk].x = qrow[d0]; aq[kk].y = qrow[d0 + 1]; }
        else      { aq[kk].x = 0.0f;     aq[kk].y = 0.0f; }
    }
    v2f ones; ones.x = 1.0f; ones.y = 1.0f;

    v8f acc[4];                  // 16 x 64 unnormalized output (4 col tiles)
    v8f accL = (v8f)0.0f;        // 16 x 16 row-sum tile (softmax denominator)
#pragma unroll
    for (int t = 0; t < 4; ++t) acc[t] = (v8f)0.0f;
    float mrow[8];               // running row max (row = r + 8*half)
#pragma unroll
    for (int r = 0; r < 8; ++r) mrow[r] = -__builtin_inff();

    int s_begin = split * SPB;
    for (int it = 0; it < SPB / SB; ++it) {
        int chunk0 = s_begin + it * SB;

        // ---- TDM: stage K/V tiles into LDS (wave 0 issues; EXEC ignored) ----
        __syncthreads();   // prior iteration done reading ktile/vtile
        if (wave == 0) {
            const float* kg = k + (((size_t)b * S_ + chunk0) * H_ + h) * D_;
            const float* vg = v + (((size_t)b * S_ + chunk0) * H_ + h) * D_;
            tdm_load_tile(koff, kg);
            tdm_load_tile(voff, vg);
            __builtin_amdgcn_s_wait_tensorcnt((short)0);
        }
        __syncthreads();   // tiles visible to all waves

        // ---- 4 score tiles: (16x64) x (64x16) each, 16 chained WMMAs --------
        v8f c[4];
#pragma unroll
        for (int ct = 0; ct < 4; ++ct) {
            const float* kb = &ktile[(ct * 16 + ln) * KSTR];
            v8f cc = (v8f)0.0f;
#pragma unroll
            for (int kk = 0; kk < 16; ++kk) {
                int d0 = kk * 4 + half * 2;
                v2f bb = *(const v2f*)&kb[d0];
                cc = wmma_f32(aq[kk], bb, cc);
            }
            c[ct] = cc;
        }

        // ---- online softmax: one shuffle-max per row over all 64 cols -------
        float fac[8];
#pragma unroll
        for (int r = 0; r < 8; ++r) {
            float s0 = fmaxf(fmaxf(c[0][r], c[1][r]), fmaxf(c[2][r], c[3][r]));
            float mx = s0 * SCALE;
#pragma unroll
            for (int off = 1; off < 16; off <<= 1)
                mx = fmaxf(mx, __shfl_xor(mx, off, 32));
            float mnew = fmaxf(mrow[r], mx);
            fac[r] = __expf(mrow[r] - mnew);
            mrow[r] = mnew;
        }
#pragma unroll
        for (int ct = 0; ct < 4; ++ct)
#pragma unroll
            for (int r = 0; r < 8; ++r)
                c[ct][r] = __expf(c[ct][r] * SCALE - mrow[r]);
#pragma unroll
        for (int t = 0; t < 4; ++t)
#pragma unroll
            for (int r = 0; r < 8; ++r) acc[t][r] *= fac[r];
#pragma unroll
        for (int r = 0; r < 8; ++r) accL[r] *= fac[r];

        // ---- stage P (16x64) through LDS: C/D layout -> A layout ------------
        __syncthreads();
#pragma unroll
        for (int ct = 0; ct < 4; ++ct)
#pragma unroll
            for (int r = 0; r < 8; ++r)
                pbuf[wave][(r + half * 8) * KSTR + ct * 16 + ln] = c[ct][r];
        __syncthreads();

        // ---- ctx: P(16x64) x V(64x64) + denominator P(16x64) x ones(64x16) --
#pragma unroll
        for (int kk = 0; kk < 16; ++kk) {
            int kd = kk * 4 + half * 2;
            v2f a = *(const v2f*)&pbuf[wave][ln * KSTR + kd];
            const float* vr0 = &vtile[kd * KSTR];
            const float* vr1 = &vtile[(kd + 1) * KSTR];
            accL = wmma_f32(a, ones, accL);
#pragma unroll
            for (int t = 0; t < 4; ++t) {
                v2f bb; bb.x = vr0[t * 16 + ln]; bb.y = vr1[t * 16 + ln];
                acc[t] = wmma_f32(a, bb, acc[t]);
            }
        }
    }

    // ---- write partials: m, l, unnormalized O -------------------------------
    size_t pb = ((size_t)bh * NSPLIT + split) * NROWS;
#pragma unroll
    for (int r = 0; r < 8; ++r) {
        int row = wave * 16 + r + half * 8;
        if (ln == 0) {
            part_m[pb + row] = mrow[r];
            part_l[pb + row] = accL[r];
        }
#pragma unroll
        for (int t = 0; t < 4; ++t)
            part_acc[(pb + row) * D_ + t * 16 + ln] = acc[t][r];
    }
}

// ---------------------------------------------------------------------------
// Kernel 6: merge split partials and scatter into out[b, mtop[n], h, :]
// ---------------------------------------------------------------------------
__global__ void probattn_combine(const float* __restrict__ part_acc,
                                 const float* __restrict__ part_m,
                                 const float* __restrict__ part_l,
                                 const int*   __restrict__ mtop,
                                 float* __restrict__ out) {
    int bh = blockIdx.x;
    int b = bh / H_, h = bh % H_;
    int d = threadIdx.x;

    for (int n = 0; n < NTOP; ++n) {
        size_t pb = (size_t)bh * NSPLIT * NROWS + n;
        float mstar = -__builtin_inff();
#pragma unroll
        for (int sp = 0; sp < NSPLIT; ++sp)
            mstar = fmaxf(mstar, part_m[pb + (size_t)sp * NROWS]);
        float Lsum = 0.0f, O = 0.0f;
#pragma unroll
        for (int sp = 0; sp < NSPLIT; ++sp) {
            float w = __expf(part_m[pb + (size_t)sp * NROWS] - mstar);
            Lsum += part_l[pb + (size_t)sp * NROWS] * w;
            O    += part_acc[(pb + (size_t)sp * NROWS) * D_ + d] * w;
        }
        int ldst = mtop[bh * NTOP + n];
        out[(((size_t)b * L_ + ldst) * H_ + h) * D_ + d] = O / Lsum;
    }
}

// ---------------------------------------------------------------------------
extern "C" void kernel_launch(void* const* d_in, const int* in_sizes, int n_in,
                              void* d_out, int out_size, void* d_ws, size_t ws_size,
                              hipStream_t stream) {
    const float* q   = (const float*)d_in[0];
    const float* k   = (const float*)d_in[1];
    const float* v   = (const float*)d_in[2];
    const int*   idx = (const int*)d_in[3];
    float* out = (float*)d_out;

    char* ws = (char*)d_ws;
    size_t off = 0;
    float* M        = (float*)(ws + off); off += (size_t)B_ * H_ * L_ * 4;          // 256 KB
    int*   mtop     = (int*)  (ws + off); off += 8192;                              // 5 KB used
    float* vmean    = (float*)(ws + off); off += (size_t)B_ * H_ * D_ * 4;          // 8 KB
    float* part_m   = (float*)(ws + off); off += (size_t)B_ * H_ * NSPLIT * NROWS * 4;
    float* part_l   = (float*)(ws + off); off += (size_t)B_ * H_ * NSPLIT * NROWS * 4;
    float* part_acc = (float*)(ws + off); // B*H*NSPLIT*NROWS*D*4 = 3 MB

    int nBHL = B_ * H_ * L_;
    probattn_measure<<<(nBHL + 127) / 128, 128, 0, stream>>>(q, k, idx, M);
    probattn_topk<<<B_ * H_, 256, 0, stream>>>(M, mtop);
    probattn_vmean<<<B_ * H_, D_, 0, stream>>>(v, vmean);
    int nOut = B_ * L_ * H_ * D_;
    probattn_bcast<<<(nOut + 255) / 256, 256, 0, stream>>>(vmean, out);
    probattn_split<<<B_ * H_ * NSPLIT, 96, 0, stream>>>(q, k, v, mtop,
                                                        part_acc, part_m, part_l);
    probattn_combine<<<B_ * H_, D_, 0, stream>>>(part_acc, part_m, part_l, mtop, out);
}